// LinearMemoryBackend_51986284151093
// MI455X (gfx1250) — compile-verified
//
#include <hip/hip_runtime.h>

// CDNA5 fp32 WMMA fragments
typedef __attribute__((ext_vector_type(2))) float v2f;   // A/B 16x4 fp32 fragment: 2 VGPRs
typedef __attribute__((ext_vector_type(8))) float v8f;   // C/D 16x16 fp32 accumulator: 8 VGPRs

#define T_SEQ 4096
#define NH    8
#define DH    64
#define CHUNK 64
#define NPART (T_SEQ / CHUNK)     // 64 partitions per head
#define NHEAD (2 * NH)            // B*H = 16
#define PITCH 68                  // multiple of 4 (float4 LDS stores), staggers banks

#define WMMA_F32(a, b, c) \
  __builtin_amdgcn_wmma_f32_16x16x4_f32(false, (a), false, (b), (short)0, (c), false, false)

// A-fragment (and transposed-B) load: element [rb+row16][kb + 2*half + {0,1}]
__device__ __forceinline__ v2f lds_fragA(const float* L, int rb, int kb, int row16, int koff) {
  v2f a;
  a.x = L[(rb + row16) * PITCH + kb + koff];
  a.y = L[(rb + row16) * PITCH + kb + koff + 1];
  return a;
}

// B-fragment (and transposed-A) load: element [kb + 2*half + {0,1}][nb+row16]
__device__ __forceinline__ v2f lds_fragB(const float* L, int kb, int nb, int row16, int koff) {
  v2f b;
  b.x = L[(kb + koff)     * PITCH + nb + row16];
  b.y = L[(kb + koff + 1) * PITCH + nb + row16];
  return b;
}

// ---------------------------------------------------------------------------
// Phase 1: per (head, partition) local state  M_p = V_p^T K_p   (64x64)
// Fully parallel: grid = NHEAD*NPART = 1024 blocks.
// ---------------------------------------------------------------------------
__global__ void __launch_bounds__(128, 1)
linmem_partial_state(const float* __restrict__ k, const float* __restrict__ v,
                     float* __restrict__ mpart) {
  __shared__ __align__(16) float Ks[CHUNK * PITCH];
  __shared__ __align__(16) float Vs[CHUNK * PITCH];

  const int tid = threadIdx.x, lane = tid & 31, wave = tid >> 5;
  const int row16 = lane & 15, half = lane >> 4;
  const int koff = half * 2, crow = half * 8;

  const int bh = blockIdx.x / NPART;
  const int p  = blockIdx.x % NPART;
  const int b = bh / NH, h = bh % NH;
  const int t0 = p * CHUNK;

  const size_t head_off = ((size_t)b * T_SEQ * NH + h) * DH;
  const size_t tstride  = (size_t)NH * DH;

  for (int idx = tid; idx < CHUNK * (DH / 4); idx += 128) {
    const int r = idx >> 4, c4 = idx & 15;
    const size_t g = head_off + (size_t)(t0 + r) * tstride + c4 * 4;
    *(float4*)&Ks[r * PITCH + c4 * 4] = *(const float4*)&k[g];
    *(float4*)&Vs[r * PITCH + c4 * 4] = *(const float4*)&v[g];
  }
  __syncthreads();

  float* out = mpart + ((size_t)bh * NPART + p) * (DH * DH);
  const int di = wave;                       // wave owns d-tile row strip
  for (int ej = 0; ej < 4; ++ej) {
    v8f acc = {};
    for (int t = 0; t < CHUNK; t += 4) {
      v2f a  = lds_fragB(Vs, t, di * 16, row16, koff);   // (V^T)[d][t] = V[t][d]
      v2f bb = lds_fragB(Ks, t, ej * 16, row16, koff);   // K[t][e]
      acc = WMMA_F32(a, bb, acc);
    }
#pragma unroll
    for (int j = 0; j < 8; ++j)
      out[(di * 16 + crow + j) * DH + ej * 16 + row16] = acc[j];
  }
}

// ---------------------------------------------------------------------------
// Phase 2: exclusive prefix-sum of partition states along p, per head.
// grid = NHEAD blocks; elementwise over the 64x64 state.
// ---------------------------------------------------------------------------
__global__ void __launch_bounds__(256, 1)
linmem_prefix_state(float* __restrict__ mpart) {
  float* base = mpart + (size_t)blockIdx.x * NPART * (DH * DH);
  for (int e = threadIdx.x; e < DH * DH; e += 256) {
    float run = 0.0f;
    for (int p = 0; p < NPART; ++p) {
      float* ptr = base + (size_t)p * (DH * DH) + e;
      const float cur = *ptr;
      *ptr = run;                 // exclusive prefix: state entering partition p
      run += cur;
    }
  }
}

// ---------------------------------------------------------------------------
// Phase 3: per (head, partition) output  Y = Q M0^T + tril(Q K^T) V
// Fully parallel: grid = NHEAD*NPART = 1024 blocks. No sequential chain.
// ---------------------------------------------------------------------------
__global__ void __launch_bounds__(128, 1)
linmem_chunk_output(const float* __restrict__ q, const float* __restrict__ k,
                    const float* __restrict__ v, const float* __restrict__ mpart,
                    float* __restrict__ y) {
  __shared__ __align__(16) float Qs[CHUNK * PITCH];
  __shared__ __align__(16) float Ks[CHUNK * PITCH];
  __shared__ __align__(16) float Vs[CHUNK * PITCH];
  __shared__ __align__(16) float Ss[CHUNK * PITCH];
  __shared__ __align__(16) float Ms[DH    * PITCH];

  const int tid = threadIdx.x, lane = tid & 31, wave = tid >> 5;
  const int row16 = lane & 15, half = lane >> 4;
  const int koff = half * 2, crow = half * 8;

  const int bh = blockIdx.x / NPART;
  const int p  = blockIdx.x % NPART;
  const int b = bh / NH, h = bh % NH;
  const int t0 = p * CHUNK;

  const size_t head_off = ((size_t)b * T_SEQ * NH + h) * DH;
  const size_t tstride  = (size_t)NH * DH;
  const float* m0 = mpart + ((size_t)bh * NPART + p) * (DH * DH);

  for (int idx = tid; idx < CHUNK * (DH / 4); idx += 128) {
    const int r = idx >> 4, c4 = idx & 15;
    const size_t g = head_off + (size_t)(t0 + r) * tstride + c4 * 4;
    *(float4*)&Qs[r * PITCH + c4 * 4] = *(const float4*)&q[g];
    *(float4*)&Ks[r * PITCH + c4 * 4] = *(const float4*)&k[g];
    *(float4*)&Vs[r * PITCH + c4 * 4] = *(const float4*)&v[g];
    *(float4*)&Ms[r * PITCH + c4 * 4] = *(const float4*)&m0[r * DH + c4 * 4];
  }
  __syncthreads();

  const int ti = wave;   // wave owns 16-row time strip: produces AND consumes its Ss rows

  // ---- S = tril(Q K^T) for this wave's row strip ----
  for (int sj = 0; sj < 4; ++sj) {
    v8f acc = {};
    if (sj <= ti) {
      for (int e = 0; e < DH; e += 4) {
        v2f a  = lds_fragA(Qs, ti * 16, e, row16, koff);
        v2f bt = lds_fragA(Ks, sj * 16, e, row16, koff);   // (K^T)[e][s] = K[s][e]
        acc = WMMA_F32(a, bt, acc);
      }
    }
#pragma unroll
    for (int j = 0; j < 8; ++j) {
      const int tt = ti * 16 + crow + j;
      const int ss = sj * 16 + row16;
      Ss[tt * PITCH + ss] = (ss <= tt) ? acc[j] : 0.0f;    // causal mask
    }
  }
  // No barrier: Ss rows [16*ti,16*ti+16) are written and read by the SAME wave;
  // same-wave LDS ops are in-order (DScnt waits inserted by compiler).

  // ---- Y = Q M0^T + S V ----
  for (int dj = 0; dj < 4; ++dj) {
    v8f acc = {};
    for (int e = 0; e < DH; e += 4) {                      // inter-partition state
      v2f a  = lds_fragA(Qs, ti * 16, e, row16, koff);
      v2f bt = lds_fragA(Ms, dj * 16, e, row16, koff);     // (M^T)[e][d] = M[d][e]
      acc = WMMA_F32(a, bt, acc);
    }
    for (int s = 0; s < CHUNK; s += 4) {                   // intra-partition causal part
      v2f a  = lds_fragA(Ss, ti * 16, s, row16, koff);
      v2f bb = lds_fragB(Vs, s, dj * 16, row16, koff);
      acc = WMMA_F32(a, bb, acc);
    }
#pragma unroll
    for (int j = 0; j < 8; ++j) {
      const size_t g = head_off +
          (size_t)(t0 + ti * 16 + crow + j) * tstride + dj * 16 + row16;
      y[g] = acc[j];
    }
  }
}

// ---------------------------------------------------------------------------
// Fallback: sequential chunked recurrence (one block per head) for the case
// where the workspace is too small for the scan decomposition.
// ---------------------------------------------------------------------------
__global__ void __launch_bounds__(128, 1)
linmem_chunked_wmma(const float* __restrict__ q,
                    const float* __restrict__ k,
                    const float* __restrict__ v,
                    float* __restrict__ y) {
  __shared__ __align__(16) float Qs[CHUNK * PITCH];
  __shared__ __align__(16) float Ks[CHUNK * PITCH];
  __shared__ __align__(16) float Vs[CHUNK * PITCH];
  __shared__ __align__(16) float Ss[CHUNK * PITCH];
  __shared__ __align__(16) float Ms[DH    * PITCH];

  const int tid = threadIdx.x, lane = tid & 31, wave = tid >> 5;
  const int row16 = lane & 15, half = lane >> 4;
  const int koff = half * 2, crow = half * 8;
  const int b = blockIdx.x / NH, h = blockIdx.x % NH;

  for (int i = tid; i < DH * PITCH; i += 128) Ms[i] = 0.0f;
  __syncthreads();

  const size_t head_off = ((size_t)b * T_SEQ * NH + h) * DH;
  const size_t tstride  = (size_t)NH * DH;

  for (int t0 = 0; t0 < T_SEQ; t0 += CHUNK) {
    for (int idx = tid; idx < CHUNK * (DH / 4); idx += 128) {
      const int r = idx >> 4, c4 = idx & 15;
      const size_t g = head_off + (size_t)(t0 + r) * tstride + c4 * 4;
      *(float4*)&Qs[r * PITCH + c4 * 4] = *(const float4*)&q[g];
      *(float4*)&Ks[r * PITCH + c4 * 4] = *(const float4*)&k[g];
      *(float4*)&Vs[r * PITCH + c4 * 4] = *(const float4*)&v[g];
      if (t0 + CHUNK < T_SEQ) {
        __builtin_prefetch(&q[g + CHUNK * tstride], 0, 3);
        __builtin_prefetch(&k[g + CHUNK * tstride], 0, 3);
        __builtin_prefetch(&v[g + CHUNK * tstride], 0, 3);
      }
    }
    __syncthreads();

    const int ti = wave;
    for (int sj = 0; sj < 4; ++sj) {
      v8f acc = {};
      if (sj <= ti) {
        for (int e = 0; e < DH; e += 4) {
          v2f a  = lds_fragA(Qs, ti * 16, e, row16, koff);
          v2f bt = lds_fragA(Ks, sj * 16, e, row16, koff);
          acc = WMMA_F32(a, bt, acc);
        }
      }
#pragma unroll
      for (int j = 0; j < 8; ++j) {
        const int tt = ti * 16 + crow + j;
        const int ss = sj * 16 + row16;
        Ss[tt * PITCH + ss] = (ss <= tt) ? acc[j] : 0.0f;
      }
    }
    __syncthreads();

    for (int dj = 0; dj < 4; ++dj) {
      v8f acc = {};
      for (int e = 0; e < DH; e += 4) {
        v2f a  = lds_fragA(Qs, ti * 16, e, row16, koff);
        v2f bt = lds_fragA(Ms, dj * 16, e, row16, koff);
        acc = WMMA_F32(a, bt, acc);
      }
      for (int s = 0; s < CHUNK; s += 4) {
        v2f a  = lds_fragA(Ss, ti * 16, s, row16, koff);
        v2f bb = lds_fragB(Vs, s, dj * 16, row16, koff);
        acc = WMMA_F32(a, bb, acc);
      }
#pragma unroll
      for (int j = 0; j < 8; ++j) {
        const size_t g = head_off +
            (size_t)(t0 + ti * 16 + crow + j) * tstride + dj * 16 + row16;
        y[g] = acc[j];
      }
    }
    __syncthreads();

    const int di = wave;
    for (int ej = 0; ej < 4; ++ej) {
      v8f acc;
#pragma unroll
      for (int j = 0; j < 8; ++j)
        acc[j] = Ms[(di * 16 + crow + j) * PITCH + ej * 16 + row16];
      for (int t = 0; t < CHUNK; t += 4) {
        v2f a  = lds_fragB(Vs, t, di * 16, row16, koff);
        v2f bb = lds_fragB(Ks, t, ej * 16, row16, koff);
        acc = WMMA_F32(a, bb, acc);
      }
#pragma unroll
      for (int j = 0; j < 8; ++j)
        Ms[(di * 16 + crow + j) * PITCH + ej * 16 + row16] = acc[j];
    }
    __syncthreads();
  }
}

extern "C" void kernel_launch(void* const* d_in, const int* in_sizes, int n_in,
                              void* d_out, int out_size, void* d_ws, size_t ws_size,
                              hipStream_t stream) {
  const float* q = (const float*)d_in[0];
  const float* k = (const float*)d_in[1];
  const float* v = (const float*)d_in[2];
  float* y = (float*)d_out;

  const size_t ws_need = (size_t)NHEAD * NPART * DH * DH * sizeof(float);  // 16 MB

  if (ws_size >= ws_need) {
    // Scan decomposition: 1024-way parallel, no sequential recurrence.
    float* mpart = (float*)d_ws;
    linmem_partial_state<<<dim3(NHEAD * NPART), dim3(128), 0, stream>>>(k, v, mpart);
    linmem_prefix_state <<<dim3(NHEAD),         dim3(256), 0, stream>>>(mpart);
    linmem_chunk_output <<<dim3(NHEAD * NPART), dim3(128), 0, stream>>>(q, k, v, mpart, y);
  } else {
    // Fallback: sequential chunk recurrence, one workgroup per head.
    linmem_chunked_wmma<<<dim3(NHEAD), dim3(128), 0, stream>>>(q, k, v, y);
  }
}